// GAT_53128745451692
// MI455X (gfx1250) — compile-verified
//
#include <hip/hip_runtime.h>
#include <hip/hip_bf16.h>

#define NN     100000
#define EE     1600000
#define NFEATC 256
#define NHIDC  32
#define NHEADC 8
#define NCLSC  16
#define ALPHAC 0.2f

typedef __attribute__((ext_vector_type(2))) float v2f;
typedef __attribute__((ext_vector_type(8))) float v8f;

__device__ __forceinline__ v8f wmma4(v2f a, v2f b, v8f c) {
  // D = A(16x4,f32) * B(4x16,f32) + C(16x16,f32)
  return __builtin_amdgcn_wmma_f32_16x16x4_f32(false, a, false, b, (short)0, c,
                                               false, false);
}

// order-preserving float -> uint encoding (for atomic segment-max)
__device__ __forceinline__ unsigned fenc(float f) {
  unsigned u = __float_as_uint(f);
  return (u & 0x80000000u) ? ~u : (u | 0x80000000u);
}
__device__ __forceinline__ float fdec(unsigned u) {
  return (u & 0x80000000u) ? __uint_as_float(u & 0x7FFFFFFFu)
                           : __uint_as_float(~u);
}
#define ENC_NEG_INF 0x007FFFFFu   // fenc(-inf)

// ---------------- fills ----------------
__global__ void fill_f32(float* p, float v, long n) {
  long i = (long)blockIdx.x * blockDim.x + threadIdx.x;
  if (i < n) p[i] = v;
}
__global__ void fill_u32(unsigned* p, unsigned v, long n) {
  long i = (long)blockIdx.x * blockDim.x + threadIdx.x;
  if (i < n) p[i] = v;
}

// ---------------- GEMM 1: Wh1[N,256] = x[N,256] @ concat(W_heads) ----------------
// block = 256 threads (8 waves), one 16-row strip per block, wave w == head w.
// A tile (16x256 f32 = 16KB) staged into LDS once per block via CDNA5 async
// copies (ASYNCcnt path), shared by all 8 waves; row pad of 2 floats avoids
// 16-way bank conflicts on the fragment reads.
#define A_STRIDE 258
__global__ void __launch_bounds__(256) gemm1_wmma(
    const float* __restrict__ x, const float* __restrict__ W,
    float* __restrict__ Wh) {
  __shared__ float As[16 * A_STRIDE];
  const int tid  = threadIdx.x;
  const int lane = tid & 31;
  const int head = tid >> 5;                    // 0..7
  const int rowBase = blockIdx.x * 16;          // N = 16*6250 exact

  // stage A: 16 rows x 64 16B-chunks = 1024 chunks; 256 threads x 4 each
  {
    int chunk = tid;
    #pragma unroll
    for (int it = 0; it < 4; ++it, chunk += 256) {
      const int ar = chunk >> 6;                // 0..15
      const int ac = (chunk & 63) << 2;         // float col, multiple of 4
      unsigned lds = (unsigned)(size_t)&As[ar * A_STRIDE + ac];
      unsigned long long ga =
          (unsigned long long)(size_t)(x + (size_t)(rowBase + ar) * NFEATC + ac);
      asm volatile("global_load_async_to_lds_b128 %0, %1, off"
                   :: "v"(lds), "v"(ga) : "memory");
    }
    asm volatile("s_wait_asynccnt 0x0" ::: "memory");
  }
  __syncthreads();

  const int r = lane & 15;
  const int koff = (lane >> 4) * 2;             // K split across half-waves
  const float* Arow  = &As[r * A_STRIDE];
  const float* Whead = W + (size_t)head * NFEATC * NHIDC;   // [256,32]
  v8f c0 = {}; v8f c1 = {};
  #pragma unroll 4
  for (int k0 = 0; k0 < NFEATC; k0 += 4) {
    v2f a; a.x = Arow[k0 + koff]; a.y = Arow[k0 + koff + 1];
    const float* wk = Whead + (size_t)(k0 + koff) * NHIDC;
    v2f b0; b0.x = wk[r];       b0.y = wk[NHIDC + r];
    v2f b1; b1.x = wk[16 + r];  b1.y = wk[NHIDC + 16 + r];
    c0 = wmma4(a, b0, c0);
    c1 = wmma4(a, b1, c1);
  }
  const int mhi = (lane >> 4) * 8;
  #pragma unroll
  for (int v = 0; v < 8; ++v) {
    size_t row = (size_t)(rowBase + mhi + v);
    Wh[row * 256 + head * NHIDC + r]      = c0[v];
    Wh[row * 256 + head * NHIDC + 16 + r] = c1[v];
  }
}

// ---------------- GEMM 2: Wh2[N,16] = h[N,256] @ W_out[256,16] ----------------
// block = 64 threads (2 waves), 3125 blocks -> exactly 6250 row tiles.
__global__ void __launch_bounds__(64) gemm2_wmma(
    const float* __restrict__ h, const float* __restrict__ Wout,
    float* __restrict__ Wh2) {
  const int lane = threadIdx.x & 31;
  const int wv   = threadIdx.x >> 5;            // 0..1
  const int rowBase = (blockIdx.x * 2 + wv) * 16;
  const int r = lane & 15;
  const int koff = (lane >> 4) * 2;
  const float* hrow = h + (size_t)(rowBase + r) * 256;
  v8f c = {};
  #pragma unroll 4
  for (int k0 = 0; k0 < 256; k0 += 4) {
    v2f a; a.x = hrow[k0 + koff]; a.y = hrow[k0 + koff + 1];
    v2f b; b.x = Wout[(size_t)(k0 + koff) * NCLSC + r];
           b.y = Wout[(size_t)(k0 + koff + 1) * NCLSC + r];
    c = wmma4(a, b, c);
  }
  const int mhi = (lane >> 4) * 8;
  #pragma unroll
  for (int v = 0; v < 8; ++v)
    Wh2[(size_t)(rowBase + mhi + v) * NCLSC + r] = c[v];
}

// ---------------- attention score projections ----------------
__global__ void __launch_bounds__(256) scores1_k(
    const float* __restrict__ Wh, const float* __restrict__ a,
    float* __restrict__ sdst, float* __restrict__ ssrc) {
  long t = (long)blockIdx.x * 256 + threadIdx.x;   // t < N*8 (exact grid)
  int n = (int)(t >> 3), head = (int)(t & 7);
  const float* w  = Wh + (size_t)n * 256 + head * NHIDC;
  const float* av = a + head * 2 * NHIDC;
  float sd = 0.f, ss = 0.f;
  #pragma unroll
  for (int f = 0; f < NHIDC; ++f) {
    float v = w[f];
    sd += v * av[f];
    ss += v * av[NHIDC + f];
  }
  sdst[t] = sd; ssrc[t] = ss;
}

__global__ void __launch_bounds__(256) scores2_k(
    const float* __restrict__ Wh2, const float* __restrict__ a,
    float* __restrict__ sdst, float* __restrict__ ssrc) {
  int n = blockIdx.x * 256 + threadIdx.x;
  if (n >= NN) return;
  float sd = 0.f, ss = 0.f;
  #pragma unroll
  for (int f = 0; f < NCLSC; ++f) {
    float v = Wh2[(size_t)n * NCLSC + f];
    sd += v * a[f];
    ss += v * a[NCLSC + f];
  }
  sdst[n] = sd; ssrc[n] = ss;
}

// ---------------- layer-1 edge passes (per edge, per head) ----------------
__global__ void __launch_bounds__(256) edge_max1(
    const int* __restrict__ edge, const float* __restrict__ sdst,
    const float* __restrict__ ssrc, unsigned* __restrict__ menc) {
  long t = (long)blockIdx.x * 256 + threadIdx.x;   // t < E*8 (exact)
  long e = t >> 3; int head = (int)(t & 7);
  int d = edge[e], s = edge[EE + e];
  float v = sdst[(size_t)d * 8 + head] + ssrc[(size_t)s * 8 + head];
  v = v > 0.f ? v : ALPHAC * v;
  atomicMax(&menc[(size_t)d * 8 + head], fenc(v));
}

__global__ void __launch_bounds__(256) edge_denom1(
    const int* __restrict__ edge, const float* __restrict__ sdst,
    const float* __restrict__ ssrc, const unsigned* __restrict__ menc,
    float* __restrict__ den) {
  long t = (long)blockIdx.x * 256 + threadIdx.x;
  long e = t >> 3; int head = (int)(t & 7);
  int d = edge[e], s = edge[EE + e];
  size_t ih = (size_t)d * 8 + head;
  float v = sdst[ih] + ssrc[(size_t)s * 8 + head];
  v = v > 0.f ? v : ALPHAC * v;
  atomicAdd(&den[ih], __expf(v - fdec(menc[ih])));
}

// one wave per (edge, head); lane = feature -> contiguous 128B gather/scatter
__global__ void __launch_bounds__(256) edge_scatter1(
    const int* __restrict__ edge, const float* __restrict__ sdst,
    const float* __restrict__ ssrc, const unsigned* __restrict__ menc,
    const float* __restrict__ den, const float* __restrict__ Wh,
    float* __restrict__ acc) {
  int e = blockIdx.x;
  int head = threadIdx.x >> 5;
  int f = threadIdx.x & 31;
  int d = edge[e], s = edge[EE + e];
  __builtin_prefetch(Wh + (size_t)s * 256 + head * NHIDC, 0, 0);
  size_t ih = (size_t)d * 8 + head;
  float v = sdst[ih] + ssrc[(size_t)s * 8 + head];
  v = v > 0.f ? v : ALPHAC * v;
  float ex = __expf(v - fdec(menc[ih]));
  float w  = ex / fmaxf(den[ih], 1e-16f);
  atomicAdd(&acc[(size_t)d * 256 + head * NHIDC + f],
            w * Wh[(size_t)s * 256 + head * NHIDC + f]);
}

__global__ void __launch_bounds__(256) elu_k(float* __restrict__ p, long n) {
  long i = (long)blockIdx.x * 256 + threadIdx.x;
  if (i < n) {
    float v = p[i];
    p[i] = v > 0.f ? v : (__expf(v) - 1.0f);
  }
}

// ---------------- output-layer edge passes (single head, F'=16) ----------------
__global__ void __launch_bounds__(256) edge_max2(
    const int* __restrict__ edge, const float* __restrict__ sdst,
    const float* __restrict__ ssrc, unsigned* __restrict__ menc) {
  long e = (long)blockIdx.x * 256 + threadIdx.x;   // E exact
  int d = edge[e], s = edge[EE + e];
  float v = sdst[d] + ssrc[s];
  v = v > 0.f ? v : ALPHAC * v;
  atomicMax(&menc[d], fenc(v));
}

__global__ void __launch_bounds__(256) edge_denom2(
    const int* __restrict__ edge, const float* __restrict__ sdst,
    const float* __restrict__ ssrc, const unsigned* __restrict__ menc,
    float* __restrict__ den) {
  long e = (long)blockIdx.x * 256 + threadIdx.x;
  int d = edge[e], s = edge[EE + e];
  float v = sdst[d] + ssrc[s];
  v = v > 0.f ? v : ALPHAC * v;
  atomicAdd(&den[d], __expf(v - fdec(menc[d])));
}

__global__ void __launch_bounds__(256) edge_scatter2(
    const int* __restrict__ edge, const float* __restrict__ sdst,
    const float* __restrict__ ssrc, const unsigned* __restrict__ menc,
    const float* __restrict__ den, const float* __restrict__ Wh2,
    float* __restrict__ out) {
  long t = (long)blockIdx.x * 256 + threadIdx.x;   // t < E*16 (exact)
  long e = t >> 4; int f = (int)(t & 15);
  int d = edge[e], s = edge[EE + e];
  float v = sdst[d] + ssrc[s];
  v = v > 0.f ? v : ALPHAC * v;
  float ex = __expf(v - fdec(menc[d]));
  float w  = ex / fmaxf(den[d], 1e-16f);
  atomicAdd(&out[(size_t)d * NCLSC + f], w * Wh2[(size_t)s * NCLSC + f]);
}

extern "C" void kernel_launch(void* const* d_in, const int* in_sizes, int n_in,
                              void* d_out, int out_size, void* d_ws, size_t ws_size,
                              hipStream_t stream) {
  const float* x       = (const float*)d_in[0];   // [N,256]
  const int*   edge    = (const int*)  d_in[1];   // [2,E]
  const float* W_heads = (const float*)d_in[2];   // [8,256,32]
  const float* a_heads = (const float*)d_in[3];   // [8,64]
  const float* W_out   = (const float*)d_in[4];   // [256,16]
  const float* a_out   = (const float*)d_in[5];   // [32]
  float* out = (float*)d_out;                     // [N,16]

  // workspace carve-up (floats)
  float* ws = (float*)d_ws;
  size_t o = 0;
  float*    Wh1   = ws + o; o += (size_t)NN * 256;
  float*    acc   = ws + o; o += (size_t)NN * 256;   // layer-1 agg -> h (in place)
  float*    sdst1 = ws + o; o += (size_t)NN * NHEADC;
  float*    ssrc1 = ws + o; o += (size_t)NN * NHEADC;
  unsigned* m1    = (unsigned*)(ws + o); o += (size_t)NN * NHEADC;
  float*    den1  = ws + o; o += (size_t)NN * NHEADC;
  float*    Wh2   = ws + o; o += (size_t)NN * NCLSC;
  float*    sdst2 = ws + o; o += NN;
  float*    ssrc2 = ws + o; o += NN;
  unsigned* m2    = (unsigned*)(ws + o); o += NN;
  float*    den2  = ws + o; o += NN;

  auto blocks = [](long n) { return (unsigned)((n + 255) / 256); };

  // init accumulators / segment-max / denominators / output
  fill_f32<<<blocks((long)NN * 256), 256, 0, stream>>>(acc, 0.f, (long)NN * 256);
  fill_u32<<<blocks((long)NN * NHEADC), 256, 0, stream>>>(m1, ENC_NEG_INF, (long)NN * NHEADC);
  fill_f32<<<blocks((long)NN * NHEADC), 256, 0, stream>>>(den1, 0.f, (long)NN * NHEADC);
  fill_u32<<<blocks(NN), 256, 0, stream>>>(m2, ENC_NEG_INF, NN);
  fill_f32<<<blocks(NN), 256, 0, stream>>>(den2, 0.f, NN);
  fill_f32<<<blocks((long)NN * NCLSC), 256, 0, stream>>>(out, 0.f, (long)NN * NCLSC);

  // ---- layer 1 ----
  gemm1_wmma<<<NN / 16, 256, 0, stream>>>(x, W_heads, Wh1);            // 6250 blocks
  scores1_k<<<(NN * NHEADC) / 256, 256, 0, stream>>>(Wh1, a_heads, sdst1, ssrc1);
  edge_max1<<<((long)EE * NHEADC) / 256, 256, 0, stream>>>(edge, sdst1, ssrc1, m1);
  edge_denom1<<<((long)EE * NHEADC) / 256, 256, 0, stream>>>(edge, sdst1, ssrc1, m1, den1);
  edge_scatter1<<<EE, 256, 0, stream>>>(edge, sdst1, ssrc1, m1, den1, Wh1, acc);
  elu_k<<<blocks((long)NN * 256), 256, 0, stream>>>(acc, (long)NN * 256);

  // ---- output layer ----
  gemm2_wmma<<<NN / 32, 64, 0, stream>>>(acc, W_out, Wh2);             // 3125 blocks
  scores2_k<<<blocks(NN), 256, 0, stream>>>(Wh2, a_out, sdst2, ssrc2);
  edge_max2<<<EE / 256, 256, 0, stream>>>(edge, sdst2, ssrc2, m2);
  edge_denom2<<<EE / 256, 256, 0, stream>>>(edge, sdst2, ssrc2, m2, den2);
  edge_scatter2<<<((long)EE * NCLSC) / 256, 256, 0, stream>>>(edge, sdst2, ssrc2, m2,
                                                              den2, Wh2, out);
}